// TensorAggregateLayer_6734508720706
// MI455X (gfx1250) — compile-verified
//
#include <hip/hip_runtime.h>
#include <math.h>

// ---------------- problem constants (from reference) ----------------
#define A_N   10000       // N_ATOMS
#define E_N   320000      // N_EDGES
#define CH    32          // channels
#define NB    8           // radial basis size
#define NC    11          // number of (in,r,out) combos
#define RCUT  5.0f
#define INV_NORM 0.03125f // 1/32
#define PI_F  3.14159265358979323846f

typedef float v2f __attribute__((ext_vector_type(2)));
typedef float v8f __attribute__((ext_vector_type(8)));

#define WPB   2           // waves per block (wave32)
#define TILE  16          // edges per wave-tile (WMMA M dimension)

// Combos (iw,rw,ow):
//  0:(0,0,0) 1:(0,1,1) 2:(0,2,2) 3:(1,0,1) 4:(1,1,0) 5:(1,1,2)
//  6:(1,2,1) 7:(2,0,2) 8:(2,1,1) 9:(2,2,0) 10:(2,2,2)

__global__ __launch_bounds__(64)
void tal_edge_kernel(const float* __restrict__ x0,
                     const float* __restrict__ x1,
                     const float* __restrict__ x2,
                     const float* __restrict__ pos,
                     const float* __restrict__ W,     // [NC][NB][CH]
                     const float* __restrict__ bias,  // [NC][CH]
                     const int*   __restrict__ idx_i,
                     const int*   __restrict__ idx_j,
                     float* __restrict__ out0,        // [A][CH]
                     float* __restrict__ out1,        // [A][CH][3]
                     float* __restrict__ out2)        // [A][CH][3][3]
{
    __shared__ float Wl[NC][NB][CH];                 // 11.0 KB
    __shared__ float bl[NC][CH];                     //  1.4 KB  (pre-scaled by 1/NORM)
    __shared__ float fnl[WPB][TILE][NC][CH];         // 44.0 KB
    __shared__ float ul[WPB][TILE][3];
    __shared__ int   iil[WPB][TILE];
    __shared__ int   jjl[WPB][TILE];

    // stage W / b once per block (bias pre-scaled by 1/NORM so it can be FMA-folded)
    for (int t = threadIdx.x; t < NC*NB*CH; t += blockDim.x) ((float*)Wl)[t] = W[t];
    for (int t = threadIdx.x; t < NC*CH;   t += blockDim.x) ((float*)bl)[t] = bias[t] * INV_NORM;
    __syncthreads();

    const int lane = threadIdx.x & 31;  // wave32
    const int wv   = threadIdx.x >> 5;
    const int hi   = lane >> 4;         // half-wave: selects K split of A/B operands
    const int lm   = lane & 15;
    const int nTiles = E_N / TILE;      // 320000/16 = 20000, exact

    for (int tile = blockIdx.x * WPB + wv; tile < nTiles; tile += gridDim.x * WPB) {
        // ---------------- phase A: geometry + radial basis ----------------
        // Both half-waves compute the same 16 edges (needed for the A-operand K split).
        const int e  = tile * TILE + lm;
        const int ii = idx_i[e];
        const int jj = idx_j[e];
        const float rx = pos[3*jj+0] - pos[3*ii+0];
        const float ry = pos[3*jj+1] - pos[3*ii+1];
        const float rz = pos[3*jj+2] - pos[3*ii+2];
        const float d    = sqrtf(rx*rx + ry*ry + rz*rz + 1e-12f);
        const float invd = 1.0f / d;
        const float u0 = rx*invd, u1 = ry*invd, u2 = rz*invd;
        const float fc   = 0.5f * (__cosf(PI_F * fminf(d, RCUT) * (1.0f/RCUT)) + 1.0f);
        const float pref = sqrtf(2.0f / RCUT) * invd * fc;
        const float td   = PI_F * d * (1.0f/RCUT);

        if (hi == 0) {
            ul[wv][lm][0] = u0; ul[wv][lm][1] = u1; ul[wv][lm][2] = u2;
            iil[wv][lm] = ii;   jjl[wv][lm] = jj;
        }

        // A operand (16x4 f32): lanes0-15 hold M=0..15 with K={0,1}, lanes16-31 K={2,3}.
        // Each lane computes only the 4 Bessel orders of its K-split half directly
        // (no lane-indexed register array -> no select ladders).
        const float kb = (float)(2*hi);              // 0 or 2
        v2f a0, a1;
        a0.x = pref * __sinf((kb + 1.0f) * td);      // K = 2*hi     -> order kb+1
        a0.y = pref * __sinf((kb + 2.0f) * td);      // K = 2*hi + 1 -> order kb+2
        a1.x = pref * __sinf((kb + 5.0f) * td);      // K = 2*hi + 4 -> order kb+5
        a1.y = pref * __sinf((kb + 6.0f) * td);      // K = 2*hi + 5 -> order kb+6

        // ---------------- phase B: fn = (rbf @ W[i] + b[i]) / NORM via WMMA ----------------
        #pragma unroll
        for (int i = 0; i < NC; ++i) {
            #pragma unroll
            for (int h = 0; h < 2; ++h) {            // channel halves N=0..15, N=16..31
                const int col = h*16 + lm;
                // B operand (4x16 f32), K split mirrors A: lanes0-15 rows {K,K+1}, lanes16-31 {K+2,K+3}
                v2f b0, b1;
                b0.x = Wl[i][2*hi + 0][col]; b0.y = Wl[i][2*hi + 1][col];
                b1.x = Wl[i][2*hi + 4][col]; b1.y = Wl[i][2*hi + 5][col];
                // accumulator starts at inline 0 (SRC2 = 0); bias+1/NORM folded into the store FMA
                v8f acc = {};
                acc = __builtin_amdgcn_wmma_f32_16x16x4_f32(false, a0, false, b0,
                                                            (short)0, acc, false, false);
                acc = __builtin_amdgcn_wmma_f32_16x16x4_f32(false, a1, false, b1,
                                                            (short)0, acc, false, false);
                const float bvn = bl[i][col];        // = b[i][col] / NORM
                // C/D layout: VGPR r holds M = r (lanes0-15) / M = r+8 (lanes16-31), N = col
                #pragma unroll
                for (int r = 0; r < 8; ++r)
                    fnl[wv][r + 8*hi][i][col] = fmaf(acc[r], INV_NORM, bvn);
            }
        }

        // LDS is in-order per wave; wait for all DS writes before cross-lane reads.
        asm volatile("s_wait_dscnt 0x0" ::: "memory");

        // ---------------- phase C: per-channel tensor algebra + atomic scatter ----------------
        const int c = lane;   // wave32 <-> 32 channels, one channel per lane
        for (int e2 = 0; e2 < TILE; ++e2) {
            const int ia = iil[wv][e2];
            const int ja = jjl[wv][e2];
            const float v0 = ul[wv][e2][0], v1 = ul[wv][e2][1], v2 = ul[wv][e2][2];
            const float* fe = &fnl[wv][e2][0][0];
            float f[NC];
            #pragma unroll
            for (int i = 0; i < NC; ++i) f[i] = fe[i*CH + c];

            const float x0v = x0[ja*CH + c];
            const float* p1 = &x1[(ja*CH + c) * 3];
            const float x1v0 = p1[0], x1v1 = p1[1], x1v2 = p1[2];
            const float* p2 = &x2[(ja*CH + c) * 9];
            float x2v[9];
            #pragma unroll
            for (int k = 0; k < 9; ++k) x2v[k] = p2[k];

            const float s1 = x1v0*v0 + x1v1*v1 + x1v2*v2;           // x1 . u
            const float y0 = x2v[0]*v0 + x2v[1]*v1 + x2v[2]*v2;     // (x2 . u)[a]
            const float y1 = x2v[3]*v0 + x2v[4]*v1 + x2v[5]*v2;
            const float y2 = x2v[6]*v0 + x2v[7]*v1 + x2v[8]*v2;
            const float s2 = y0*v0 + y1*v1 + y2*v2;                 // u . x2 . u

            // out0: combos 0,4,9
            atomicAdd(&out0[ia*CH + c], f[0]*x0v + f[4]*s1 + f[9]*s2);

            // out1: combos 1,3,6,8
            const float g = f[1]*x0v + f[6]*s1;
            float* o1 = &out1[(ia*CH + c) * 3];
            atomicAdd(&o1[0], g*v0 + f[3]*x1v0 + f[8]*y0);
            atomicAdd(&o1[1], g*v1 + f[3]*x1v1 + f[8]*y1);
            atomicAdd(&o1[2], g*v2 + f[3]*x1v2 + f[8]*y2);

            // out2: combos 2,5,7,10 ; out2[a][b] += h[a]*u[b] + f7*x2[a][b]
            const float f2x = f[2]*x0v;
            const float h0 = f2x*v0 + f[5]*x1v0 + f[10]*y0;
            const float h1 = f2x*v1 + f[5]*x1v1 + f[10]*y1;
            const float h2 = f2x*v2 + f[5]*x1v2 + f[10]*y2;
            float* o2 = &out2[(ia*CH + c) * 9];
            atomicAdd(&o2[0], h0*v0 + f[7]*x2v[0]);
            atomicAdd(&o2[1], h0*v1 + f[7]*x2v[1]);
            atomicAdd(&o2[2], h0*v2 + f[7]*x2v[2]);
            atomicAdd(&o2[3], h1*v0 + f[7]*x2v[3]);
            atomicAdd(&o2[4], h1*v1 + f[7]*x2v[4]);
            atomicAdd(&o2[5], h1*v2 + f[7]*x2v[5]);
            atomicAdd(&o2[6], h2*v0 + f[7]*x2v[6]);
            atomicAdd(&o2[7], h2*v1 + f[7]*x2v[7]);
            atomicAdd(&o2[8], h2*v2 + f[7]*x2v[8]);
        }
        // keep next tile's LDS writes from being hoisted above this tile's reads
        asm volatile("" ::: "memory");
    }
}

extern "C" void kernel_launch(void* const* d_in, const int* in_sizes, int n_in,
                              void* d_out, int out_size, void* d_ws, size_t ws_size,
                              hipStream_t stream) {
    const float* x0    = (const float*)d_in[0];
    const float* x1    = (const float*)d_in[1];
    const float* x2    = (const float*)d_in[2];
    const float* pos   = (const float*)d_in[3];
    const float* W     = (const float*)d_in[4];
    const float* b     = (const float*)d_in[5];
    const int*   idx_i = (const int*)d_in[6];
    const int*   idx_j = (const int*)d_in[7];

    float* out0 = (float*)d_out;                 // [A,CH]
    float* out1 = out0 + (size_t)A_N * CH;       // [A,CH,3]
    float* out2 = out1 + (size_t)A_N * CH * 3;   // [A,CH,3,3]

    // zero outputs (harness poisons d_out); memsetAsync is graph-capturable
    hipMemsetAsync(d_out, 0, (size_t)out_size * sizeof(float), stream);

    tal_edge_kernel<<<dim3(2048), dim3(64), 0, stream>>>(
        x0, x1, x2, pos, W, b, idx_i, idx_j, out0, out1, out2);
}